// RNN_41558103556411
// MI455X (gfx1250) — compile-verified
//
#include <hip/hip_runtime.h>

// ---------------------------------------------------------------------------
// Elman RNN for MI455X (gfx1250), fp32 via V_WMMA_F32_16X16X4_F32.
//   B=64, S=512, I=1024, H=1024
//   Phase 1: x_proj = seq @ w_ih^T + b_ih  (32768x1024x1024 GEMM, WMMA f32)
//   Phase 2: persistent kernel: 32 WGs, each WG's 32-row w_hh strip staged
//            into LDS once by TDM (tensor_load_to_lds), grid barrier between
//            the 512 steps, hidden state ping-ponged TRANSPOSED (H x B)
//            through d_ws so A-fragment loads are fully coalesced.
//   Fallback (tiny d_ws): per-step launches.
// ---------------------------------------------------------------------------

typedef float v2f __attribute__((ext_vector_type(2)));
typedef float v8f __attribute__((ext_vector_type(8)));
typedef unsigned int v4u __attribute__((ext_vector_type(4)));
typedef int v4i __attribute__((ext_vector_type(4)));
typedef int v8i __attribute__((ext_vector_type(8)));

#define RNN_B 64
#define RNN_S 512
#define RNN_I 1024
#define RNN_H 1024

// LDS row stride for the w_hh tile: TDM pads 2 DWORDs after every 256 DWORDs
// -> 1024 + 4*2 = 1032 floats per row. Keeps ds_load_b64 8B-aligned and the
// 16 B-fragment lanes spread across banks (stride%64 == 8, 2-way conflict).
#define LDSW_STRIDE 1032

__device__ __forceinline__ v8f wmma_f32(v2f a, v2f b, v8f c) {
    // 8 args: (neg_a, A, neg_b, B, c_mod, C, reuse_a, reuse_b)
    return __builtin_amdgcn_wmma_f32_16x16x4_f32(false, a, false, b,
                                                 (short)0, c, false, false);
}

// ---------------------------------------------------------------------------
// TDM: load `rows` consecutive rows of a row-major (rows x 1024) f32 tensor
// from global into LDS at byte offset lds_byte, padding 2 DWORDs per 256.
// ---------------------------------------------------------------------------
__device__ __forceinline__ void tdm_load_rows(const float* gsrc,
                                              unsigned lds_byte, int rows)
{
    unsigned long long ga = (unsigned long long)(uintptr_t)gsrc;

    v4u g0;
    g0.x = 1u;                                   // count=1 (valid), user mode
    g0.y = lds_byte;                             // lds_addr
    g0.z = (unsigned)(ga & 0xFFFFFFFFu);         // global_addr[31:0]
    g0.w = (unsigned)((ga >> 32) & 0x01FFFFFFu)  // global_addr[56:32]
         | (2u << 30);                           // type = 2 (image)

    v8i g1;
    g1[0] = (int)((2u << 16)      // data_size = 4B
                | (1u << 20)      // pad_enable
                | (7u << 22)      // pad_interval: 256 DWORDs
                | (1u << 25));    // pad_amount:   2 DWORDs
    g1[1] = (int)(1024u << 16);                 // tensor_dim0 = 1024
    g1[2] = (int)((unsigned)rows << 16);        // tensor_dim1 = rows
    g1[3] = (int)(1024u << 16);                 // tile_dim0 = 1024
    g1[4] = rows;                               // tile_dim1 = rows, tile_dim2=0
    g1[5] = 1024;                               // tensor_dim0_stride = 1024
    g1[6] = 0;
    g1[7] = 0;

    v4i gz = {0, 0, 0, 0};
#if __clang_major__ >= 23
    v8i gz8 = {0, 0, 0, 0, 0, 0, 0, 0};
    __builtin_amdgcn_tensor_load_to_lds(g0, g1, gz, gz, gz8, 0);
#else
    __builtin_amdgcn_tensor_load_to_lds(g0, g1, gz, gz, 0);
#endif
}

// ---------------------------------------------------------------------------
// Grid-wide sense/generation barrier (deterministic; integer atomics only).
// bar[0] = arrival count, bar[1] = generation. Zeroed by hipMemsetAsync.
// ---------------------------------------------------------------------------
__device__ __forceinline__ void grid_sync(unsigned* bar, unsigned nblk)
{
    __threadfence();      // make this thread's global writes device-visible
    __syncthreads();
    if (threadIdx.x == 0) {
        unsigned g = __hip_atomic_load(&bar[1], __ATOMIC_RELAXED,
                                       __HIP_MEMORY_SCOPE_AGENT);
        unsigned a = __hip_atomic_fetch_add(&bar[0], 1u, __ATOMIC_ACQ_REL,
                                            __HIP_MEMORY_SCOPE_AGENT);
        if (a == nblk - 1) {
            __hip_atomic_store(&bar[0], 0u, __ATOMIC_RELAXED,
                               __HIP_MEMORY_SCOPE_AGENT);
            __hip_atomic_store(&bar[1], g + 1u, __ATOMIC_RELEASE,
                               __HIP_MEMORY_SCOPE_AGENT);
        } else {
            while (__hip_atomic_load(&bar[1], __ATOMIC_ACQUIRE,
                                     __HIP_MEMORY_SCOPE_AGENT) == g) {
                __builtin_amdgcn_s_sleep(2);
            }
        }
    }
    __syncthreads();
}

// ---------------------------------------------------------------------------
// Phase 1: out[m,n] = sum_k seq[m,k]*w_ih[n,k] + b_ih[n]
// Wave = 16x64 C strip (4 WMMA tiles share one A fragment); 8 waves/block.
// ---------------------------------------------------------------------------
__global__ __launch_bounds__(256)
void rnn_xproj_kernel(const float* __restrict__ seq,
                      const float* __restrict__ w_ih,
                      const float* __restrict__ b_ih,
                      float* __restrict__ out)
{
    const int lane = threadIdx.x & 31;
    const int wave = threadIdx.x >> 5;
    const int half = lane >> 4;
    const int l16  = lane & 15;

    const int m0 = blockIdx.x * 128 + wave * 16;
    const int n0 = blockIdx.y * 64;

    const float* aptr = seq  + (size_t)(m0 + l16) * RNN_I + half * 2;
    const float* bptr = w_ih + (size_t)(n0 + l16) * RNN_I + half * 2;

    v8f c0 = {}, c1 = {}, c2 = {}, c3 = {};

    for (int kb = 0; kb < RNN_I; kb += 64) {
        __builtin_prefetch(aptr + kb + 64, 0, 3);
        __builtin_prefetch(bptr + kb + 64, 0, 3);
        __builtin_prefetch(bptr + 16 * RNN_I + kb + 64, 0, 3);
        __builtin_prefetch(bptr + 32 * RNN_I + kb + 64, 0, 3);
        __builtin_prefetch(bptr + 48 * RNN_I + kb + 64, 0, 3);
#pragma unroll
        for (int k = kb; k < kb + 64; k += 4) {
            v2f a  = *(const v2f*)(aptr + k);
            v2f b0 = *(const v2f*)(bptr + k);
            v2f b1 = *(const v2f*)(bptr + 16 * RNN_I + k);
            v2f b2 = *(const v2f*)(bptr + 32 * RNN_I + k);
            v2f b3 = *(const v2f*)(bptr + 48 * RNN_I + k);
            c0 = wmma_f32(a, b0, c0);
            c1 = wmma_f32(a, b1, c1);
            c2 = wmma_f32(a, b2, c2);
            c3 = wmma_f32(a, b3, c3);
        }
    }

    const float bias0 = b_ih[n0 +  0 + l16];
    const float bias1 = b_ih[n0 + 16 + l16];
    const float bias2 = b_ih[n0 + 32 + l16];
    const float bias3 = b_ih[n0 + 48 + l16];

    const int mb = m0 + half * 8;
#pragma unroll
    for (int j = 0; j < 8; ++j) {
        size_t r = (size_t)(mb + j) * RNN_H + n0 + l16;
        out[r +  0] = c0[j] + bias0;
        out[r + 16] = c1[j] + bias1;
        out[r + 32] = c2[j] + bias2;
        out[r + 48] = c3[j] + bias3;
    }
}

// ---------------------------------------------------------------------------
// Phase 2 (persistent): 32 blocks x 128 threads (4 waves), dynamic LDS.
// Block owns w_hh rows [n0, n0+32) staged in LDS by 4 parallel TDM ops
// (8 rows each).  Wave w computes a 16x32 strip (batch rows m0 = 16w, two
// 16x16 WMMA tiles sharing each A fragment), full K=1024 per step.
// Hidden state ping-pongs TRANSPOSED (hT[k*64+m]) between hb0/hb1 so the
// A-fragment loads are coalesced b32s.  4 independent accumulator chains.
// ---------------------------------------------------------------------------
__global__ __launch_bounds__(128)
void rnn_persist_kernel(const float* __restrict__ h0,
                        const float* __restrict__ w_hh,
                        const float* __restrict__ b_hh,
                        float* __restrict__ out,
                        float* __restrict__ hb0,   // hT ping
                        float* __restrict__ hb1,   // hT pong
                        unsigned* __restrict__ bar)
{
    extern __shared__ float lds_w[];      // 32 * LDSW_STRIDE floats (132 KB)

    const int lane = threadIdx.x & 31;
    const int wave = threadIdx.x >> 5;    // 0..3
    const int half = lane >> 4;
    const int l16  = lane & 15;
    const int n0   = blockIdx.x * 32;     // hidden strip (gridDim.x = 32)
    const int m0   = wave * 16;           // batch tile

    // --- stage w_hh[n0:n0+32, :] into LDS: 4 TDM loads, 8 rows per wave ---
    {
        unsigned lds_base = (unsigned)(uintptr_t)(&lds_w[0]);
        tdm_load_rows(w_hh + (size_t)(n0 + wave * 8) * RNN_H,
                      lds_base + (unsigned)(wave * 8 * LDSW_STRIDE * 4), 8);
        __builtin_amdgcn_s_wait_tensorcnt(0);
    }

    // --- transpose-copy h0 strip into hb0: hb0[n*64+m] = h0[m*H+n] ---
    for (int idx = threadIdx.x; idx < 32 * RNN_B; idx += 128) {
        int nl = idx >> 6;                 // 0..31
        int m  = idx & (RNN_B - 1);        // 0..63
        hb0[(size_t)(n0 + nl) * RNN_B + m] = h0[(size_t)m * RNN_H + n0 + nl];
    }
    __syncthreads();
    grid_sync(bar, 32);

    const float bias0 = b_hh[n0 +  0 + l16];
    const float bias1 = b_hh[n0 + 16 + l16];
    const int   SH    = RNN_S * RNN_H;

    // A fragment element k lives at aptr[k*64]; .y component at +64
    const int aoff = half * 2 * RNN_B + m0 + l16;

    for (int s = 0; s < RNN_S; ++s) {
        const float* rb = (s & 1) ? hb1 : hb0;
        float*       wb = (s & 1) ? hb0 : hb1;

        v8f c00 = {}, c01 = {}, c10 = {}, c11 = {};
        for (int kb = 0; kb < RNN_H; kb += 256) {
            const float* abase = rb + (size_t)kb * RNN_B + aoff;
            const int    lbase = half * 2 + kb + 2 * (kb >> 8);  // TDM padding
            const float* l0 = &lds_w[(l16     ) * LDSW_STRIDE + lbase];
            const float* l1 = &lds_w[(l16 + 16) * LDSW_STRIDE + lbase];
#pragma unroll
            for (int k2 = 0; k2 < 256; k2 += 8) {
                v2f a0, a1;
                a0.x = abase[(k2 + 0) * RNN_B];
                a0.y = abase[(k2 + 1) * RNN_B];
                a1.x = abase[(k2 + 4) * RNN_B];
                a1.y = abase[(k2 + 5) * RNN_B];
                v2f b00 = *(const v2f*)(l0 + k2);
                v2f b10 = *(const v2f*)(l1 + k2);
                v2f b01 = *(const v2f*)(l0 + k2 + 4);
                v2f b11 = *(const v2f*)(l1 + k2 + 4);
                c00 = wmma_f32(a0, b00, c00);
                c10 = wmma_f32(a0, b10, c10);
                c01 = wmma_f32(a1, b01, c01);
                c11 = wmma_f32(a1, b11, c11);
            }
        }
        v8f c0 = c00 + c01;
        v8f c1 = c10 + c11;

        float* xrow = out + (size_t)s * RNN_H;   // row (b) stride = S*H
        const int mb = m0 + half * 8;
#pragma unroll
        for (int j = 0; j < 8; ++j) {
            int    m   = mb + j;
            size_t gi0 = (size_t)m * SH + n0 + l16;
            float  v0  = xrow[gi0]      + c0[j] + bias0;
            float  v1  = xrow[gi0 + 16] + c1[j] + bias1;
            v0 = v0 > 0.0f ? v0 : 0.0f;
            v1 = v1 > 0.0f ? v1 : 0.0f;
            xrow[gi0]      = v0;                               // output h_s
            xrow[gi0 + 16] = v1;
            wb[(size_t)(n0 +      l16) * RNN_B + m] = v0;      // hT next step
            wb[(size_t)(n0 + 16 + l16) * RNN_B + m] = v1;
        }
        grid_sync(bar, 32);
    }
}

// ---------------------------------------------------------------------------
// Phase 2 fallback (d_ws too small): one launch per step, LDS K-reduction.
// ---------------------------------------------------------------------------
__global__ __launch_bounds__(128)
void rnn_step_kernel(const float* __restrict__ hprev, int hstride,
                     const float* __restrict__ w_hh,
                     const float* __restrict__ b_hh,
                     float* __restrict__ out_s)
{
    const int SH   = RNN_S * RNN_H;
    const int lane = threadIdx.x & 31;
    const int wave = threadIdx.x >> 5;
    const int half = lane >> 4;
    const int l16  = lane & 15;

    const int m0 = blockIdx.x * 16;
    const int n0 = blockIdx.y * 32;
    const int k0 = wave * 256;

    const float* aptr = hprev + (size_t)(m0 + l16) * hstride + k0 + half * 2;
    const float* bptr = w_hh  + (size_t)(n0 + l16) * RNN_H   + k0 + half * 2;

    v8f c0 = {}, c1 = {};
#pragma unroll 16
    for (int k = 0; k < 256; k += 4) {
        v2f a  = *(const v2f*)(aptr + k);
        v2f b0 = *(const v2f*)(bptr + k);
        v2f b1 = *(const v2f*)(bptr + 16 * RNN_H + k);
        c0 = wmma_f32(a, b0, c0);
        c1 = wmma_f32(a, b1, c1);
    }

    __shared__ float red[3][2][8][32];
    if (wave > 0) {
#pragma unroll
        for (int j = 0; j < 8; ++j) {
            red[wave - 1][0][j][lane] = c0[j];
            red[wave - 1][1][j][lane] = c1[j];
        }
    }
    __syncthreads();

    if (wave == 0) {
#pragma unroll
        for (int w = 0; w < 3; ++w)
#pragma unroll
            for (int j = 0; j < 8; ++j) {
                c0[j] += red[w][0][j][lane];
                c1[j] += red[w][1][j][lane];
            }

        const float bias0 = b_hh[n0 +  0 + l16];
        const float bias1 = b_hh[n0 + 16 + l16];
        const int mb = m0 + half * 8;
#pragma unroll
        for (int j = 0; j < 8; ++j) {
            size_t r = (size_t)(mb + j) * SH + n0 + l16;
            float x0 = out_s[r +  0] + c0[j] + bias0;
            float x1 = out_s[r + 16] + c1[j] + bias1;
            out_s[r +  0] = x0 > 0.0f ? x0 : 0.0f;
            out_s[r + 16] = x1 > 0.0f ? x1 : 0.0f;
        }
    }
}

// h_last[b,h] = out[b, S-1, h]
__global__ __launch_bounds__(256)
void rnn_hlast_kernel(const float* __restrict__ out, float* __restrict__ hlast)
{
    int i = blockIdx.x * 256 + threadIdx.x;
    int b = i >> 10;
    int h = i & (RNN_H - 1);
    hlast[i] = out[(size_t)b * RNN_S * RNN_H + (size_t)(RNN_S - 1) * RNN_H + h];
}

extern "C" void kernel_launch(void* const* d_in, const int* in_sizes, int n_in,
                              void* d_out, int out_size, void* d_ws, size_t ws_size,
                              hipStream_t stream)
{
    (void)in_sizes; (void)n_in; (void)out_size;

    const float* seq  = (const float*)d_in[0];   // (B, S, I)
    const float* h0   = (const float*)d_in[1];   // (B, H)
    const float* w_ih = (const float*)d_in[2];   // (H, I)
    const float* w_hh = (const float*)d_in[3];   // (H, H)
    const float* b_ih = (const float*)d_in[4];   // (H)
    const float* b_hh = (const float*)d_in[5];   // (H)
    float* out = (float*)d_out;                  // (B,S,H) then (B,H)

    // Phase 1: input-projection GEMM into out (x_proj storage)
    dim3 g1((RNN_B * RNN_S) / 128, RNN_H / 64);  // 256 x 16
    rnn_xproj_kernel<<<g1, 256, 0, stream>>>(seq, w_ih, b_ih, out);

    const size_t hbytes = (size_t)RNN_B * RNN_H * sizeof(float);   // 256 KB
    const size_t need   = 2 * hbytes + 64;

    if (ws_size >= need) {
        // Persistent recurrence: single kernel, grid barrier between steps.
        float*    hb0 = (float*)d_ws;
        float*    hb1 = (float*)((char*)d_ws + hbytes);
        unsigned* bar = (unsigned*)((char*)d_ws + 2 * hbytes);
        hipMemsetAsync(bar, 0, 2 * sizeof(unsigned), stream);
        const size_t lds_bytes = (size_t)32 * LDSW_STRIDE * sizeof(float);
        rnn_persist_kernel<<<32, 128, lds_bytes, stream>>>(h0, w_hh, b_hh, out,
                                                           hb0, hb1, bar);
    } else {
        // Fallback: one launch per timestep.
        dim3 g2(RNN_B / 16, RNN_H / 32);
        for (int s = 0; s < RNN_S; ++s) {
            const float* hprev = (s == 0) ? h0 : (out + (size_t)(s - 1) * RNN_H);
            int hstride        = (s == 0) ? RNN_H : RNN_S * RNN_H;
            rnn_step_kernel<<<g2, 128, 0, stream>>>(hprev, hstride, w_hh, b_hh,
                                                    out + (size_t)s * RNN_H);
        }
    }

    rnn_hlast_kernel<<<(RNN_B * RNN_H) / 256, 256, 0, stream>>>(
        out, out + (size_t)RNN_B * RNN_S * RNN_H);
}